// AFM_6700148981883
// MI455X (gfx1250) — compile-verified
//
#include <hip/hip_runtime.h>

// AFM forward, algebraically reduced:
//   attention is softmax over a size-1 axis => weights == 1.0 exactly,
//   Sum_{i<j} e_i*e_j = 0.5*((Sum_f e_f)^2 - Sum_f e_f^2)  (elementwise)
//   out[b] = sigmoid(out_k * Sum_e pooled[b,e] + out_b)
//
// The field reduction is executed on the CDNA5 matrix pipe as
// D[b,e] = Sum_k M[b,k] * Bsel[k,e], k = f*16+e', Bsel = block identity,
// using V_WMMA_F32_16X16X4_F32 (fp32 end-to-end; S^2-Q cancellation needs it).

#define N_BATCH 4096
#define NF      26
#define VOCAB_N 100000
#define EMB_N   16

typedef float v2f __attribute__((ext_vector_type(2)));
typedef float v8f __attribute__((ext_vector_type(8)));

__global__ __launch_bounds__(256) void afm_fused_kernel(
    const int*   __restrict__ sparse_ids,   // [B, 26] int32
    const float* __restrict__ tables,       // [26, VOCAB, 16] f32
    const float* __restrict__ out_kernel,   // [1,1]
    const float* __restrict__ out_bias,     // [1]
    float*       __restrict__ out)          // [B, 1]
{
    __shared__ float xpose[8][16][16];      // 8 KB: per-wave pooled tile

    const int tid  = threadIdx.x;
    const int lane = tid & 31;
    const int wave = tid >> 5;
    const int kh   = lane >> 4;             // half-wave: selects K pair {0,1} vs {2,3}
    const int ln   = lane & 15;             // row (M=b) index within tile / N=e index
    const int b0   = (blockIdx.x * 8 + wave) * 16;
    const int brow = b0 + ln;               // both half-waves work on batch row b0+ln

    // Preload this row's 26 sparse ids (stays in VGPRs).
    int ids[NF];
#pragma unroll
    for (int f = 0; f < NF; ++f) ids[f] = sparse_ids[brow * NF + f];

    // Constant B operands: 4 chunks of the 16x16 identity, 4 K-rows each.
    // B layout (4x16 f32): VGPR0 = rows K=0 (lanes 0-15) / K=2 (lanes 16-31),
    //                      VGPR1 = rows K=1 / K=3  (mirrors the A-matrix K split).
    v2f bsel[4];
#pragma unroll
    for (int s = 0; s < 4; ++s) {
        bsel[s].x = (ln == 4 * s + 2 * kh)     ? 1.0f : 0.0f;
        bsel[s].y = (ln == 4 * s + 1 + 2 * kh) ? 1.0f : 0.0f;
    }

    v8f accS = {};   // S[b,e] = Sum_f emb[b,f,e]
    v8f accQ = {};   // Q[b,e] = Sum_f emb[b,f,e]^2

    for (int f = 0; f < NF; ++f) {
        const float* row =
            tables + ((size_t)f * VOCAB_N + (size_t)(unsigned)ids[f]) * EMB_N;
        if (f + 2 < NF) {
            const float* nrow =
                tables + ((size_t)(f + 2) * VOCAB_N + (size_t)(unsigned)ids[f + 2]) * EMB_N;
            __builtin_prefetch(nrow, 0, 1);          // global_prefetch_b8
        }
#pragma unroll
        for (int s = 0; s < 4; ++s) {
            // A-matrix 16x4 f32 layout: lane ln (half 0) holds {K=0,K=1},
            // lane ln (half 1) holds {K=2,K=3} of row M=ln -> one b64 gather.
            v2f a = *(const v2f*)(row + 4 * s + 2 * kh);
            accS = __builtin_amdgcn_wmma_f32_16x16x4_f32(
                       false, a, false, bsel[s], (short)0, accS, false, false);
            v2f a2 = { a.x * a.x, a.y * a.y };
            accQ = __builtin_amdgcn_wmma_f32_16x16x4_f32(
                       false, a2, false, bsel[s], (short)0, accQ, false, false);
        }
    }

    // C/D layout: VGPR r, lanes 0-15 -> (M=r, N=ln); lanes 16-31 -> (M=8+r, N=ln).
    // pooled[b,e] = 0.5*(S^2 - Q); stash per-wave tile in LDS for the e-reduction.
#pragma unroll
    for (int r = 0; r < 8; ++r) {
        float sv = accS[r];
        float qv = accQ[r];
        xpose[wave][r + 8 * kh][ln] = 0.5f * (sv * sv - qv);
    }
    __syncthreads();

    if (kh == 0) {
        float x = 0.0f;
#pragma unroll
        for (int e = 0; e < EMB_N; ++e) x += xpose[wave][ln][e];
        float t = x * out_kernel[0] + out_bias[0];
        out[b0 + ln] = 1.0f / (1.0f + __expf(-t));
    }
}

extern "C" void kernel_launch(void* const* d_in, const int* in_sizes, int n_in,
                              void* d_out, int out_size, void* d_ws, size_t ws_size,
                              hipStream_t stream) {
    (void)in_sizes; (void)n_in; (void)out_size; (void)d_ws; (void)ws_size;
    // setup_inputs order:
    // 0 dense_x (unused), 1 sparse_ids, 2 embed_tables,
    // 3..6 attention params (dead code: softmax over size-1 axis == 1),
    // 7 out_kernel, 8 out_bias
    const int*   ids    = (const int*)d_in[1];
    const float* tables = (const float*)d_in[2];
    const float* okern  = (const float*)d_in[7];
    const float* obias  = (const float*)d_in[8];
    float*       out    = (float*)d_out;

    dim3 grid(N_BATCH / (8 * 16));   // 32 blocks
    dim3 block(256);                 // 8 wave32 waves, 16 batch rows each
    afm_fused_kernel<<<grid, block, 0, stream>>>(ids, tables, okern, obias, out);
}